// BiliSample_56590489092844
// MI455X (gfx1250) — compile-verified
//
#include <hip/hip_runtime.h>

typedef __attribute__((ext_vector_type(2))) float v2f;
typedef __attribute__((ext_vector_type(8))) float v8f;

#define Zd    200
#define Yd    8
#define Xd    200
#define Wimg  64      // actual feature-map W (img_feat.shape[3])
#define Himg  32      // actual feature-map H (img_feat.shape[2])
#define Cfeat 64
#define Kdim  512     // Cfeat * Yd
#define OUTC  128
#define MROWS 64      // M rows per block (4 WMMA subtiles of 16)
#define LDS_STRIDE 516  // 512 + 4 pad: bank-conflict-free b64 A-fragment reads

__global__ __launch_bounds__(256) void fused_bev_wmma(
    const float* __restrict__ img,   // (2,64,32,64)
    const float* __restrict__ P,     // (2,3,3)
    const float* __restrict__ Wmat,  // (128,512) row-major [o][k]
    const float* __restrict__ bias,  // (128)
    float* __restrict__ out)         // (2,128,200,200)
{
  extern __shared__ float Atile[];           // MROWS * LDS_STRIDE floats (132,096 B)

  const int tid  = threadIdx.x;
  const int tile = blockIdx.x;               // 0..1249 ; 64 spatial points each

  // ---------- Stage 1: bilinear sampling -> LDS A-tile [64 m][512 k], k = c*8 + y
  // 64 m * 8 y = 512 locations; thread -> (location, channel-half), 4 passes.
  #pragma unroll
  for (int iter = 0; iter < 4; ++iter) {
    const int loc = (tid >> 1) + iter * 128; // 0..511 = (m, y)
    const int m   = loc >> 3;                // 0..63
    const int y   = loc & 7;                 // 0..7
    const int c0  = (tid & 1) * 32;          // channel half

    const int p = tile * MROWS + m;          // flat (b,z,x) in [0, 80000)
    const int x = p % Xd;
    const int z = (p / Xd) % Zd;
    const int b = p / (Xd * Zd);

    const float fx = ((float)x * (1.0f / (float)(Xd - 1)) - 0.5f) * 80.0f;
    const float fy = ((float)y * (1.0f / (float)(Yd - 1)) - 0.5f) * 6.0f;
    const float fz = ((float)z * (1.0f / (float)(Zd - 1))) * 80.0f;

    const float* Pb = P + b * 9;
    const float px = Pb[0] * fx + Pb[1] * fy + Pb[2] * fz;
    const float py = Pb[3] * fx + Pb[4] * fy + Pb[5] * fz;
    const float d  = Pb[6] * fx + Pb[7] * fy + Pb[8] * fz;

    const float inv = 1.0f / (d + 1e-8f);
    const float u01 = px * inv * (1.0f / 512.0f);   // IMG_W
    const float v01 = py * inv * (1.0f / 256.0f);   // IMG_H
    const float validf =
        (u01 >= 0.0f && v01 >= 0.0f && u01 <= 1.0f && v01 <= 1.0f) ? 1.0f : 0.0f;

    const float xp  = u01 * (float)Wimg - 0.5f;
    const float yp  = v01 * (float)Himg - 0.5f;
    const float x0f = floorf(xp), y0f = floorf(yp);
    const float wx1 = xp - x0f,   wy1 = yp - y0f;
    const int x0 = (int)x0f, y0 = (int)y0f;
    const int x1 = x0 + 1,   y1 = y0 + 1;

    const bool okx0 = (x0 >= 0) && (x0 < Wimg);
    const bool okx1 = (x1 >= 0) && (x1 < Wimg);
    const bool oky0 = (y0 >= 0) && (y0 < Himg);
    const bool oky1 = (y1 >= 0) && (y1 < Himg);
    const int xc0 = min(max(x0, 0), Wimg - 1);
    const int xc1 = min(max(x1, 0), Wimg - 1);
    const int yc0 = min(max(y0, 0), Himg - 1);
    const int yc1 = min(max(y1, 0), Himg - 1);

    const float w00 = (1.0f - wx1) * (1.0f - wy1) * ((okx0 && oky0) ? 1.0f : 0.0f) * validf;
    const float w01 = wx1 * (1.0f - wy1)          * ((okx1 && oky0) ? 1.0f : 0.0f) * validf;
    const float w10 = (1.0f - wx1) * wy1          * ((okx0 && oky1) ? 1.0f : 0.0f) * validf;
    const float w11 = wx1 * wy1                   * ((okx1 && oky1) ? 1.0f : 0.0f) * validf;

    const float* base = img + (size_t)(b * Cfeat + c0) * (Himg * Wimg);
    const int o00 = yc0 * Wimg + xc0;
    const int o01 = yc0 * Wimg + xc1;
    const int o10 = yc1 * Wimg + xc0;
    const int o11 = yc1 * Wimg + xc1;

    float* dst = Atile + m * LDS_STRIDE + c0 * 8 + y;
    #pragma unroll 8
    for (int c = 0; c < 32; ++c) {
      const float* ch = base + c * (Himg * Wimg);
      dst[c * 8] = w00 * ch[o00] + w01 * ch[o01] + w10 * ch[o10] + w11 * ch[o11];
    }
  }
  __syncthreads();

  // ---------- Stage 2: WMMA f32 GEMM. Wave wv: D[64 m][16 o] as 4 subtiles,
  // one shared B fragment per K-step (4x global-load amortization).
  const int lane    = tid & 31;
  const int wv      = tid >> 5;        // 0..7 -> o-tile
  const int halfsel = lane >> 4;       // 0: K pair (0,1)   1: K pair (2,3)
  const int lrow    = lane & 15;       // A: M row / B: N col

  // A 16x4 f32 layout: lanes 0-15 hold {K0,K1} of row M=lane; lanes 16-31 {K2,K3}
  const float* ldsA = Atile + lrow * LDS_STRIDE + halfsel * 2;
  // B 4x16 f32 layout: lanes 0-15 hold {B[0,n],B[1,n]}; lanes 16-31 {B[2,n],B[3,n]}
  // B[k,n] = conv_w[wv*16+n, k] -> contiguous pair in row-major conv_w
  const float* gB = Wmat + (size_t)(wv * 16 + lrow) * Kdim + halfsel * 2;

  v8f acc[4];
  #pragma unroll
  for (int mt = 0; mt < 4; ++mt)
    acc[mt] = (v8f){0.f, 0.f, 0.f, 0.f, 0.f, 0.f, 0.f, 0.f};

  #pragma unroll 4
  for (int kk = 0; kk < Kdim; kk += 4) {
    v2f bb;
    bb.x = gB[kk];
    bb.y = gB[kk + 1];
    #pragma unroll
    for (int mt = 0; mt < 4; ++mt) {
      v2f a;
      a.x = ldsA[mt * 16 * LDS_STRIDE + kk];
      a.y = ldsA[mt * 16 * LDS_STRIDE + kk + 1];
      // 8 args: (neg_a, A, neg_b, B, c_mod, C, reuse_a, reuse_b)
      acc[mt] = __builtin_amdgcn_wmma_f32_16x16x4_f32(false, a, false, bb,
                                                      (short)0, acc[mt], false, false);
    }
  }

  // ---------- Stage 3: bias + store. D: VGPR v -> M=v (lanes 0-15) / v+8 (16-31).
  const int o  = wv * 16 + lrow;
  const float bo = bias[o];
  #pragma unroll
  for (int mt = 0; mt < 4; ++mt) {
    #pragma unroll
    for (int v = 0; v < 8; ++v) {
      const int m = mt * 16 + v + halfsel * 8;
      const int p = tile * MROWS + m;
      const int x = p % Xd;
      const int z = (p / Xd) % Zd;
      const int b = p / (Xd * Zd);
      out[(((size_t)b * OUTC + o) * Zd + z) * Xd + x] = acc[mt][v] + bo;
    }
  }
}

extern "C" void kernel_launch(void* const* d_in, const int* in_sizes, int n_in,
                              void* d_out, int out_size, void* d_ws, size_t ws_size,
                              hipStream_t stream) {
  const float* img  = (const float*)d_in[0];   // img_feat  (2,64,32,64)
  const float* P    = (const float*)d_in[1];   // pix_T_cam (2,3,3)
  const float* Wm   = (const float*)d_in[2];   // conv_w    (128,512)
  const float* bs   = (const float*)d_in[3];   // conv_b    (128)
  float* out = (float*)d_out;                  // (2,128,200,200)

  const size_t shmem = (size_t)MROWS * LDS_STRIDE * sizeof(float);  // 132,096 B
  (void)hipFuncSetAttribute((const void*)fused_bev_wmma,
                            hipFuncAttributeMaxDynamicSharedMemorySize, (int)shmem);

  // 80,000 spatial points / 64 per block = 1250 blocks, 256 threads (8 waves)
  fused_bev_wmma<<<1250, 256, shmem, stream>>>(img, P, Wm, bs, out);
}